// OneHotWeightedAverage_71330816852664
// MI455X (gfx1250) — compile-verified
//
#include <hip/hip_runtime.h>

#define B 256
#define V 50257
#define GLOBAL_AS __attribute__((address_space(1)))
#define LDS_AS    __attribute__((address_space(3)))

#if __has_builtin(__builtin_amdgcn_global_load_async_to_lds_b32) && \
    __has_builtin(__builtin_amdgcn_s_wait_asynccnt)
#define USE_ASYNC 1
#else
#define USE_ASYNC 0
#endif

__device__ __forceinline__ float wave_max32(float v) {
#pragma unroll
  for (int off = 16; off > 0; off >>= 1)
    v = fmaxf(v, __shfl_xor(v, off, 32));
  return v;
}
__device__ __forceinline__ float wave_sum32(float v) {
#pragma unroll
  for (int off = 16; off > 0; off >>= 1)
    v += __shfl_xor(v, off, 32);
  return v;
}

// ---------------------------------------------------------------------------
// Kernel A: per-row softmax statistics. One 1024-thread block (32 waves) per
// row. Pass 1: row max. Pass 2: sum of exp(x - max) (second pass hits L2:
// one row = 201 KB << 192 MB). Writes rowmax[b], rowinvsum[b] to workspace.
// ---------------------------------------------------------------------------
__global__ void __launch_bounds__(1024)
row_stats_kernel(const float* __restrict__ w, float* __restrict__ rmax,
                 float* __restrict__ rinv) {
  const int b = blockIdx.x;
  const int tid = threadIdx.x;
  const int lane = tid & 31, wid = tid >> 5;
  __shared__ float red[32];
  const float* row = w + (size_t)b * V;

  float m = -3.402823466e38f;
  for (int v = tid; v < V; v += 1024) {
    __builtin_prefetch(&row[v + 8192], 0, 1);   // global_prefetch_b8
    m = fmaxf(m, row[v]);
  }
  m = wave_max32(m);
  if (lane == 0) red[wid] = m;
  __syncthreads();
  if (wid == 0) {
    float t = red[lane];
    t = wave_max32(t);
    if (lane == 0) red[0] = t;
  }
  __syncthreads();
  const float M = red[0];
  __syncthreads();  // red[] reuse below

  float s = 0.0f;
  for (int v = tid; v < V; v += 1024)
    s += __expf(row[v] - M);
  s = wave_sum32(s);
  if (lane == 0) red[wid] = s;
  __syncthreads();
  if (wid == 0) {
    float t = red[lane];
    t = wave_sum32(t);
    if (lane == 0) {
      rmax[b] = M;
      rinv[b] = 1.0f / t;
    }
  }
}

// ---------------------------------------------------------------------------
// Kernel B: per-row scatter-add, privatized in LDS. CDNA5's 320 KB LDS/WGP
// holds the entire 201 KB vocab row, so the scatter uses ds_add_f32 LDS
// atomics (no global atomics, no global zero pass), then streams the row out
// fully coalesced.
// ---------------------------------------------------------------------------
__global__ void __launch_bounds__(1024)
scatter_average_kernel(const int* __restrict__ idx, const float* __restrict__ w,
                       const float* __restrict__ rmax,
                       const float* __restrict__ rinv,
                       float* __restrict__ avg) {
  extern __shared__ float hist[];  // V floats = 201,028 bytes (dynamic LDS)
  const int b = blockIdx.x;
  const int tid = threadIdx.x;

  for (int v = tid; v < V; v += 1024) hist[v] = 0.0f;
  __syncthreads();

  const float M = rmax[b];
  const float inv = rinv[b];
  const float* row = w + (size_t)b * V;
  const int* ind = idx + (size_t)b * V;

  for (int v = tid; v < V; v += 1024) {
    __builtin_prefetch(&ind[v + 8192], 0, 1);
    const float wt = __expf(row[v] - M) * inv;
    atomicAdd(&hist[ind[v]], wt);  // -> ds_add_f32
  }
  __syncthreads();

  float* out = avg + (size_t)b * V;
  for (int v = tid; v < V; v += 1024) out[v] = hist[v];
}

// ---------------------------------------------------------------------------
// Kernel C: weights.T [V, B]. 32x32 tiles staged in LDS (pitch 33: coprime
// with the 64 banks -> conflict-free transposed reads). Tile loads use the
// gfx1250 async global->LDS path (ASYNCcnt + s_wait_asynccnt); stores are
// coalesced 128B wave-rows along B.
// ---------------------------------------------------------------------------
__global__ void __launch_bounds__(1024)
transpose_weights_kernel(const float* __restrict__ w,
                         const float* __restrict__ rmax,
                         const float* __restrict__ rinv,
                         float* __restrict__ wT) {
  __shared__ float tile[32][33];
  const int tx = threadIdx.x, ty = threadIdx.y;
  const int v0 = blockIdx.x * 32;
  const int b0 = blockIdx.y * 32;

  const int vload = v0 + tx;
  if (vload < V) {
    const float* src = w + (size_t)(b0 + ty) * V + vload;
#if USE_ASYNC
    __builtin_amdgcn_global_load_async_to_lds_b32(
        (GLOBAL_AS int*)src, (LDS_AS int*)&tile[ty][tx], 0, 0);
#else
    tile[ty][tx] = *src;
#endif
  }
#if USE_ASYNC
  __builtin_amdgcn_s_wait_asynccnt(0);
#endif
  __syncthreads();

  const int vstore = v0 + ty;
  if (vstore < V) {
    const int b = b0 + tx;                       // lane-contiguous in B
    const float x = tile[tx][ty];                // = w_es[b][vstore]
    const float wt = __expf(x - rmax[b]) * rinv[b];
    wT[(size_t)vstore * B + b] = wt;             // coalesced 128B per row
  }
}

extern "C" void kernel_launch(void* const* d_in, const int* in_sizes, int n_in,
                              void* d_out, int out_size, void* d_ws,
                              size_t ws_size, hipStream_t stream) {
  (void)in_sizes; (void)n_in; (void)out_size; (void)ws_size;
  const int* indices = (const int*)d_in[0];
  const float* w_es = (const float*)d_in[1];
  float* out = (float*)d_out;
  float* avg = out;                       // [B, V]
  float* wT = out + (size_t)B * V;        // [V, B]
  float* rmax = (float*)d_ws;             // [B]
  float* rinv = rmax + B;                 // [B]

  // Raise dynamic-LDS cap for the 201 KB histogram (CDNA5: 320 KB per WGP).
  // Host-side attribute set; non-enqueuing, deterministic, capture-safe.
  (void)hipFuncSetAttribute((const void*)scatter_average_kernel,
                            hipFuncAttributeMaxDynamicSharedMemorySize,
                            (int)(V * sizeof(float)));

  row_stats_kernel<<<B, 1024, 0, stream>>>(w_es, rmax, rinv);
  scatter_average_kernel<<<B, 1024, V * sizeof(float), stream>>>(
      indices, w_es, rmax, rinv, avg);
  dim3 grid((V + 31) / 32, B / 32);
  dim3 blk(32, 32);
  transpose_weights_kernel<<<grid, blk, 0, stream>>>(w_es, rmax, rinv, wT);
}